// CDD_82806969467444
// MI455X (gfx1250) — compile-verified
//
#include <hip/hip_runtime.h>
#include <hip/hip_bf16.h>

#define N_USER 80000
#define N_ITEM 20000
#define NN     (N_USER + N_ITEM)   // 100000
#define EE     1600000
#define DD     128
#define LL     3
#define KK     3
#define BB     4096
#define OUTCOLS (4 * DD)           // 512

typedef __attribute__((ext_vector_type(2))) float v2f;
typedef __attribute__((ext_vector_type(8))) float v8f;

// ---------------------------------------------------------------------------
// Build ego = [[user_emb, feat]; [item_emb1, item_emb2]]  (N x 128)
// and copy it into all_emb columns [0,128).
// ---------------------------------------------------------------------------
__global__ void build_ego(const float* __restrict__ feat,
                          const float* __restrict__ user_emb,
                          const float* __restrict__ item_emb1,
                          const float* __restrict__ item_emb2,
                          float* __restrict__ ego,
                          float* __restrict__ all_emb) {
    int idx = blockIdx.x * blockDim.x + threadIdx.x;
    if (idx >= NN * DD) return;
    int n = idx / DD;
    int d = idx - n * DD;
    float v;
    if (d < 64) {
        v = (n < N_USER) ? user_emb[n * 64 + d]
                         : item_emb1[(n - N_USER) * 64 + d];
    } else {
        int dd = d - 64;
        v = (n < N_USER) ? feat[n * 64 + dd]
                         : item_emb2[(n - N_USER) * 64 + dd];
    }
    ego[idx] = v;
    all_emb[(size_t)n * OUTCOLS + d] = v;
}

// ---------------------------------------------------------------------------
// betas[k][i] = softmax(alphas[k]) , L rows of K
// ---------------------------------------------------------------------------
__global__ void softmax_beta(const float* __restrict__ alphas,
                             float* __restrict__ betas) {
    int k = threadIdx.x;
    if (k >= LL) return;
    float m = -1e30f;
    for (int i = 0; i < KK; ++i) m = fmaxf(m, alphas[k * KK + i]);
    float e[KK];
    float s = 0.f;
    for (int i = 0; i < KK; ++i) { e[i] = __expf(alphas[k * KK + i] - m); s += e[i]; }
    float inv = 1.0f / s;
    for (int i = 0; i < KK; ++i) betas[k * KK + i] = e[i] * inv;
}

// ---------------------------------------------------------------------------
// Zero a float4 buffer
// ---------------------------------------------------------------------------
__global__ void zero_buf(float4* __restrict__ p, int n4) {
    int i = blockIdx.x * blockDim.x + threadIdx.x;
    if (i < n4) p[i] = make_float4(0.f, 0.f, 0.f, 0.f);
}

// ---------------------------------------------------------------------------
// SpMM via edge-parallel atomic scatter: Y[row] += val * X[col].
// One wave (32 lanes) per edge; each lane owns 4 consecutive columns.
// Working set fits in the 192MB L2 -> atomics resolve in-cache.
// ---------------------------------------------------------------------------
__global__ __launch_bounds__(256)
void spmm_atomic(const int* __restrict__ erow, const int* __restrict__ ecol,
                 const float* __restrict__ eval,
                 const float* __restrict__ X, float* __restrict__ Y) {
    int wave = (blockIdx.x * blockDim.x + threadIdx.x) >> 5;
    int lane = threadIdx.x & 31;
    if (wave >= EE) return;
    int   r = erow[wave];
    int   c = ecol[wave];
    float v = eval[wave];
    const float4* xp = (const float4*)(X + (size_t)c * DD);
    float4 x = xp[lane];
    float* yp = Y + (size_t)r * DD + lane * 4;
    unsafeAtomicAdd(yp + 0, v * x.x);
    unsafeAtomicAdd(yp + 1, v * x.y);
    unsafeAtomicAdd(yp + 2, v * x.z);
    unsafeAtomicAdd(yp + 3, v * x.w);
}

// ---------------------------------------------------------------------------
// side = (init ? 0 : side) + beta * cur
// ---------------------------------------------------------------------------
__global__ void axpy_side(const float* __restrict__ cur, float* __restrict__ side,
                          const float* __restrict__ betas, int bidx, int init) {
    int i = blockIdx.x * blockDim.x + threadIdx.x;
    if (i >= NN * DD / 4) return;
    float b = betas[bidx];
    float4 c = ((const float4*)cur)[i];
    float4 s = init ? make_float4(0.f, 0.f, 0.f, 0.f) : ((const float4*)side)[i];
    s.x += b * c.x; s.y += b * c.y; s.z += b * c.z; s.w += b * c.w;
    ((float4*)side)[i] = s;
}

// ---------------------------------------------------------------------------
// Fused layer transform using V_WMMA_F32_16X16X4_F32:
//   ego <- leaky_relu(side @ Wgc + bgc, 0.2) + (ego .* side) @ Wbi + bbi
// Block = 256 threads = 8 waves; block owns 16 rows; wave w owns cols
// [16w, 16w+16). K-loop of 32 steps, 2 WMMA per step (gc and bi paths).
// A fragments come from LDS; B fragments from global (W is 64KB, L2-hot).
// ---------------------------------------------------------------------------
__global__ __launch_bounds__(256)
void layer_gemm(const float* __restrict__ side, float* __restrict__ ego,
                const float* __restrict__ Wgc, const float* __restrict__ bgc,
                const float* __restrict__ Wbi, const float* __restrict__ bbi) {
    __shared__ float s_side[16 * DD];
    __shared__ float s_prod[16 * DD];
    int rowbase = blockIdx.x * 16;
    int tid = threadIdx.x;

    // Stage 16x128 of side and ego.*side into LDS (float4 per thread x2).
    const float4* sp = (const float4*)(side + (size_t)rowbase * DD);
    const float4* ep = (const float4*)(ego + (size_t)rowbase * DD);
    for (int i = tid; i < 16 * DD / 4; i += 256) {
        float4 sv = sp[i];
        float4 ev = ep[i];
        ((float4*)s_side)[i] = sv;
        ((float4*)s_prod)[i] = make_float4(sv.x * ev.x, sv.y * ev.y,
                                           sv.z * ev.z, sv.w * ev.w);
    }
    __syncthreads();

    int wave = tid >> 5;
    int lane = tid & 31;
    int half = lane >> 4;      // 0: lanes 0-15, 1: lanes 16-31
    int m    = lane & 15;
    int colbase = wave * 16;
    int col = colbase + m;

    v8f acc_gc = {};
    v8f acc_bi = {};
    for (int k0 = 0; k0 < DD; k0 += 4) {
        int ka = k0 + 2 * half;                  // this lane's K pair
        v2f a_s = *(const v2f*)(s_side + m * DD + ka);
        v2f a_p = *(const v2f*)(s_prod + m * DD + ka);
        v2f b_g, b_b;
        b_g.x = Wgc[(size_t)ka * DD + col];
        b_g.y = Wgc[(size_t)(ka + 1) * DD + col];
        b_b.x = Wbi[(size_t)ka * DD + col];
        b_b.y = Wbi[(size_t)(ka + 1) * DD + col];
        acc_gc = __builtin_amdgcn_wmma_f32_16x16x4_f32(
            false, a_s, false, b_g, (short)0, acc_gc, false, false);
        acc_bi = __builtin_amdgcn_wmma_f32_16x16x4_f32(
            false, a_p, false, b_b, (short)0, acc_bi, false, false);
    }

    float bg = bgc[col];
    float bb = bbi[col];
    // C/D layout: VGPR r holds row (r + 8*half), column col.
    for (int r = 0; r < 8; ++r) {
        float s = acc_gc[r] + bg;
        s = (s > 0.f) ? s : 0.2f * s;            // leaky_relu(0.2)
        float o = s + acc_bi[r] + bb;
        ego[(size_t)(rowbase + r + 8 * half) * DD + col] = o;
    }
}

// ---------------------------------------------------------------------------
// Row L2-normalize ego (N x 128) into all_emb columns [colOff, colOff+128).
// One wave per row; 4 floats per lane; wave32 shfl_xor reduction.
// ---------------------------------------------------------------------------
__global__ void normalize_rows(const float* __restrict__ ego,
                               float* __restrict__ all_emb, int colOff) {
    int row  = blockIdx.x * (blockDim.x >> 5) + (threadIdx.x >> 5);
    int lane = threadIdx.x & 31;
    if (row >= NN) return;
    float4 x = ((const float4*)(ego + (size_t)row * DD))[lane];
    float s = x.x * x.x + x.y * x.y + x.z * x.z + x.w * x.w;
    for (int off = 16; off > 0; off >>= 1) s += __shfl_xor(s, off, 32);
    float n = sqrtf(s);
    float scale = 1.0f / fmaxf(n, 1e-12f);
    float4 o = make_float4(x.x * scale, x.y * scale, x.z * scale, x.w * scale);
    ((float4*)(all_emb + (size_t)row * OUTCOLS + colOff))[lane] = o;
}

// ---------------------------------------------------------------------------
// Gather output rows: [u_emb[users]; i_emb[pos]; i_emb[neg]] each B x 512
// ---------------------------------------------------------------------------
__global__ void gather_out(const float* __restrict__ all_emb,
                           const int* __restrict__ users,
                           const int* __restrict__ pos,
                           const int* __restrict__ neg,
                           float* __restrict__ out) {
    int b = blockIdx.x;                 // 0 .. 3*BB-1
    int which = b / BB;
    int idx = b - which * BB;
    int row;
    if (which == 0)      row = users[idx];
    else if (which == 1) row = N_USER + pos[idx];
    else                 row = N_USER + neg[idx];
    const float4* src = (const float4*)(all_emb + (size_t)row * OUTCOLS);
    float4* dst = (float4*)(out + (size_t)b * OUTCOLS);
    for (int i = threadIdx.x; i < OUTCOLS / 4; i += blockDim.x)
        dst[i] = src[i];
}

// ---------------------------------------------------------------------------
extern "C" void kernel_launch(void* const* d_in, const int* in_sizes, int n_in,
                              void* d_out, int out_size, void* d_ws, size_t ws_size,
                              hipStream_t stream) {
    const int*   erow      = (const int*)d_in[0];
    const int*   ecol      = (const int*)d_in[1];
    const float* eval      = (const float*)d_in[2];
    const float* feat      = (const float*)d_in[3];
    const float* user_emb  = (const float*)d_in[4];
    const float* item_emb1 = (const float*)d_in[5];
    const float* item_emb2 = (const float*)d_in[6];
    const float* alphas    = (const float*)d_in[7];
    const float* W_gc      = (const float*)d_in[8];
    const float* b_gc      = (const float*)d_in[9];
    const float* W_bi      = (const float*)d_in[10];
    const float* b_bi      = (const float*)d_in[11];
    const int*   users     = (const int*)d_in[12];
    const int*   pos       = (const int*)d_in[13];
    const int*   neg       = (const int*)d_in[14];
    float* out = (float*)d_out;

    // Workspace layout (floats): all_emb | ego | side | curA | curB | betas
    float* ws = (float*)d_ws;
    size_t off = 0;
    float* all_emb = ws + off; off += (size_t)NN * OUTCOLS;
    float* ego     = ws + off; off += (size_t)NN * DD;
    float* side    = ws + off; off += (size_t)NN * DD;
    float* curA    = ws + off; off += (size_t)NN * DD;
    float* curB    = ws + off; off += (size_t)NN * DD;
    float* betas   = ws + off; off += LL * KK;

    {
        int total = NN * DD;
        build_ego<<<(total + 255) / 256, 256, 0, stream>>>(
            feat, user_emb, item_emb1, item_emb2, ego, all_emb);
    }
    softmax_beta<<<1, 32, 0, stream>>>(alphas, betas);

    const int spmm_blocks = (EE + 7) / 8;          // 8 edges per 256-thread block
    const int nd4  = NN * DD / 4;
    const int axpy_blocks = (nd4 + 255) / 256;

    for (int k = 0; k < LL; ++k) {
        const float* src = ego;
        float* cur = curA;
        float* nxt = curB;
        for (int i = 0; i < KK; ++i) {
            zero_buf<<<axpy_blocks, 256, 0, stream>>>((float4*)cur, nd4);
            spmm_atomic<<<spmm_blocks, 256, 0, stream>>>(erow, ecol, eval, src, cur);
            axpy_side<<<axpy_blocks, 256, 0, stream>>>(cur, side, betas, k * KK + i, i == 0);
            src = cur;
            float* t = cur; cur = nxt; nxt = t;    // ping-pong
        }
        layer_gemm<<<NN / 16, 256, 0, stream>>>(
            side, ego,
            W_gc + (size_t)k * DD * DD, b_gc + (size_t)k * DD,
            W_bi + (size_t)k * DD * DD, b_bi + (size_t)k * DD);
        normalize_rows<<<(NN + 7) / 8, 256, 0, stream>>>(ego, all_emb, (k + 1) * DD);
    }

    gather_out<<<3 * BB, 128, 0, stream>>>(all_emb, users, pos, neg, out);
}